// CustomMaskRCNN_67113158967561
// MI455X (gfx1250) — compile-verified
//
#include <hip/hip_runtime.h>
#include <hip/hip_bf16.h>

#define NPROP   1000
#define NGT     200
#define IMG_H   520
#define IMG_W   704
#define MS      28
#define MS2     (MS * MS)
#define IOU_THR 0.3f

typedef __attribute__((ext_vector_type(2))) float v2f;
typedef __attribute__((ext_vector_type(8))) float v8f;

typedef __attribute__((address_space(1))) int* gint_p;
typedef __attribute__((address_space(3))) int* lint_p;

#ifndef __has_builtin
#define __has_builtin(x) 0
#endif

#if __has_builtin(__builtin_amdgcn_global_load_async_to_lds_b32)
#define USE_ASYNC_LDS 1
#else
#define USE_ASYNC_LDS 0
#endif

// ---------------------------------------------------------------------------
// Kernel 1: IoU matrix row-max / argmax.  gt_boxes (200x4 f32 = 3.2 KB) staged
// into LDS via the CDNA5 async global->LDS path when available.
// ---------------------------------------------------------------------------
__global__ __launch_bounds__(256) void iou_match_kernel(
    const float* __restrict__ proposals, const float* __restrict__ gt_boxes,
    int* __restrict__ midx, float* __restrict__ pos) {
  __shared__ float sb[NGT * 4];
  const int tid = threadIdx.x;

#if USE_ASYNC_LDS
  for (int i = tid; i < NGT * 4; i += 256) {
    float* gsrc = const_cast<float*>(gt_boxes + i);
    __builtin_amdgcn_global_load_async_to_lds_b32(
        (gint_p)gsrc, (lint_p)(&sb[i]), /*offset=*/0, /*cpol=*/0);
  }
  asm volatile("s_wait_asynccnt 0" ::: "memory");
#else
  for (int i = tid; i < NGT * 4; i += 256) sb[i] = gt_boxes[i];
#endif
  __syncthreads();

  const int n = blockIdx.x * 256 + tid;
  if (n >= NPROP) return;

  const float ax1 = proposals[n * 4 + 0];
  const float ay1 = proposals[n * 4 + 1];
  const float ax2 = proposals[n * 4 + 2];
  const float ay2 = proposals[n * 4 + 3];
  const float area_a = (ax2 - ax1) * (ay2 - ay1);

  float best = -1.0f;
  int bidx = 0;
  for (int g = 0; g < NGT; ++g) {
    const float bx1 = sb[g * 4 + 0];
    const float by1 = sb[g * 4 + 1];
    const float bx2 = sb[g * 4 + 2];
    const float by2 = sb[g * 4 + 3];
    const float area_b = (bx2 - bx1) * (by2 - by1);
    const float ltx = fmaxf(ax1, bx1);
    const float lty = fmaxf(ay1, by1);
    const float rbx = fminf(ax2, bx2);
    const float rby = fminf(ay2, by2);
    const float w = fmaxf(rbx - ltx, 0.0f);
    const float h = fmaxf(rby - lty, 0.0f);
    const float inter = w * h;
    const float iou = inter / (area_a + area_b - inter);
    if (iou > best) { best = iou; bidx = g; }  // strict '>' == argmax-first tie rule
  }
  midx[n] = bidx;
  pos[n] = (best > IOU_THR) ? 1.0f : 0.0f;
}

// ---------------------------------------------------------------------------
// Kernel 2: per-proposal bilinear mask-target extraction + BCE, fixed-order
// in-block LDS tree reduction -> ws_bce[n] = pos[n] * sum_{28x28} bce.
// ---------------------------------------------------------------------------
__global__ __launch_bounds__(256) void mask_bce_kernel(
    const float* __restrict__ mask_logits, const float* __restrict__ gt_boxes,
    const float* __restrict__ gt_masks, const int* __restrict__ midx,
    const float* __restrict__ pos, float* __restrict__ ws_bce) {
  const int n = blockIdx.x;
  const int tid = threadIdx.x;

  const int m = midx[n];
  const float* box = gt_boxes + (size_t)m * 4;
  // trunc toward zero then clamp, per reference
  int x1 = (int)truncf(box[0]);
  int y1 = (int)truncf(box[1]);
  int x2 = (int)truncf(box[2]);
  int y2 = (int)truncf(box[3]);
  x1 = min(max(x1, 0), IMG_W - 1);
  y1 = min(max(y1, 0), IMG_H - 1);
  x2 = max(x1 + 1, min(x2, IMG_W));
  y2 = max(y1 + 1, min(y2, IMG_H));
  const float cw = (float)(x2 - x1);
  const float ch = (float)(y2 - y1);
  const int xmax = x2 - x1 - 1;
  const int ymax = y2 - y1 - 1;

  const float* __restrict__ mask = gt_masks + (size_t)m * (IMG_H * IMG_W);
  const float* __restrict__ zrow = mask_logits + ((size_t)n * 2 + 1) * MS2;

  // Warm the crop region's first row into cache (gfx1250 global_prefetch_b8).
  __builtin_prefetch(mask + (size_t)y1 * IMG_W + x1, 0, 0);

  float acc = 0.0f;
  for (int p = tid; p < MS2; p += 256) {
    const int i = p / MS;
    const int j = p % MS;
    const float sx = fmaxf(((float)j + 0.5f) * (1.0f / MS) * cw - 0.5f, 0.0f);
    const float sy = fmaxf(((float)i + 0.5f) * (1.0f / MS) * ch - 0.5f, 0.0f);
    const float x0f = floorf(sx);
    const float y0f = floorf(sy);
    const float wx = sx - x0f;
    const float wy = sy - y0f;
    const int x0i = (int)x0f;
    const int y0i = (int)y0f;
    const int x1i = min(x0i + 1, xmax);
    const int y1i = min(y0i + 1, ymax);
    const int gx0 = x1 + x0i, gx1 = x1 + x1i;
    const int gy0 = y1 + y0i, gy1 = y1 + y1i;
    const float v00 = mask[(size_t)gy0 * IMG_W + gx0];
    const float v01 = mask[(size_t)gy0 * IMG_W + gx1];
    const float v10 = mask[(size_t)gy1 * IMG_W + gx0];
    const float v11 = mask[(size_t)gy1 * IMG_W + gx1];
    const float top = v00 * (1.0f - wx) + v01 * wx;
    const float bot = v10 * (1.0f - wx) + v11 * wx;
    const float t = top * (1.0f - wy) + bot * wy;
    const float z = zrow[p];
    acc += fmaxf(z, 0.0f) - z * t + log1pf(expf(-fabsf(z)));
  }

  __shared__ float red[256];
  red[tid] = acc;
  __syncthreads();
  for (int s = 128; s > 0; s >>= 1) {
    if (tid < s) red[tid] += red[tid + s];
    __syncthreads();
  }
  if (tid == 0) ws_bce[n] = red[0] * pos[n];
}

// ---------------------------------------------------------------------------
// Kernel 3: single-wave deterministic reduction via V_WMMA_F32_16X16X4_F32.
// Lane partials go in A (VGPR0 = p[lane], VGPR1 = 0); B = ones =>
// D[m][n] = p[m] + p[m+16].  Sum of accumulator VGPRs on lanes 0 and 16 gives
// the exact f32 total of all 32 partials.
// ---------------------------------------------------------------------------
__global__ __launch_bounds__(32) void finalize_kernel(
    const float* __restrict__ ws_bce, const float* __restrict__ ws_pos,
    float* __restrict__ out, int n) {
  const int lane = threadIdx.x;
  float pb = 0.0f, pp = 0.0f;
  for (int i = lane; i < n; i += 32) {
    pb += ws_bce[i];
    pp += ws_pos[i];
  }

  v2f a_b; a_b[0] = pb; a_b[1] = 0.0f;
  v2f a_p; a_p[0] = pp; a_p[1] = 0.0f;
  v2f ones; ones[0] = 1.0f; ones[1] = 1.0f;
  v8f c = {};
  // 8 args: (neg_a, A, neg_b, B, c_mod, C, reuse_a, reuse_b)
  v8f db = __builtin_amdgcn_wmma_f32_16x16x4_f32(false, a_b, false, ones,
                                                 (short)0, c, false, false);
  v8f dp = __builtin_amdgcn_wmma_f32_16x16x4_f32(false, a_p, false, ones,
                                                 (short)0, c, false, false);
  float sb = 0.0f, sp = 0.0f;
#pragma unroll
  for (int v = 0; v < 8; ++v) { sb += db[v]; sp += dp[v]; }
  // lane 0 holds rows 0..7, lane 16 holds rows 8..15
  const float sum_bce = __shfl(sb, 0, 32) + __shfl(sb, 16, 32);
  const float num_pos = __shfl(sp, 0, 32) + __shfl(sp, 16, 32);

  if (lane == 0) {
    const float denom = fmaxf(num_pos, 1.0f) * (float)MS2;
    const float loss = sum_bce / denom;
    out[0] = (num_pos > 0.0f) ? loss : 0.0f;
  }
}

// ---------------------------------------------------------------------------
extern "C" void kernel_launch(void* const* d_in, const int* in_sizes, int n_in,
                              void* d_out, int out_size, void* d_ws, size_t ws_size,
                              hipStream_t stream) {
  (void)in_sizes; (void)n_in; (void)out_size; (void)ws_size;
  const float* mask_logits = (const float*)d_in[0];  // (1000, 2, 28, 28)
  const float* proposals   = (const float*)d_in[1];  // (1000, 4)
  const float* gt_boxes    = (const float*)d_in[2];  // (200, 4)
  const float* gt_masks    = (const float*)d_in[3];  // (200, 520, 704)
  float* out = (float*)d_out;

  char* ws = (char*)d_ws;
  int*   ws_midx = (int*)(ws + 0);       // 1000 ints
  float* ws_pos  = (float*)(ws + 4096);  // 1000 floats
  float* ws_bce  = (float*)(ws + 8192);  // 1000 floats

  iou_match_kernel<<<dim3((NPROP + 255) / 256), dim3(256), 0, stream>>>(
      proposals, gt_boxes, ws_midx, ws_pos);
  mask_bce_kernel<<<dim3(NPROP), dim3(256), 0, stream>>>(
      mask_logits, gt_boxes, gt_masks, ws_midx, ws_pos, ws_bce);
  finalize_kernel<<<dim3(1), dim3(32), 0, stream>>>(ws_bce, ws_pos, out, NPROP);
}